// VanillaGNN_58557584113801
// MI455X (gfx1250) — compile-verified
//
#include <hip/hip_runtime.h>

typedef __attribute__((ext_vector_type(2))) float v2f;
typedef __attribute__((ext_vector_type(8))) float v8f;
typedef __attribute__((ext_vector_type(4))) unsigned int v4u;
typedef __attribute__((ext_vector_type(8))) int v8i;
typedef __attribute__((ext_vector_type(4))) int v4i;

#define N_NODES 100000
#define IN_CH   128
#define HID_CH  128
#define OUT_CH  64
#define N_EDGES 1600000

// ---------------------------------------------------------------------------
// Zero-fill (float4 grid-stride). Counts here are multiples of 4.
// ---------------------------------------------------------------------------
__global__ __launch_bounds__(256) void gnn_zero_kernel(float4* __restrict__ p, int n4) {
    int i = blockIdx.x * blockDim.x + threadIdx.x;
    const int stride = gridDim.x * blockDim.x;
    const float4 z = make_float4(0.f, 0.f, 0.f, 0.f);
    for (; i < n4; i += stride) p[i] = z;
}

__device__ __forceinline__ void atomic_add_f32(float* p, float v) {
    unsafeAtomicAdd(p, v);   // hardware global_atomic_add_f32 (no CAS loop)
}

// ---------------------------------------------------------------------------
// GEMM1: h1[100000,128] = x[100000,128] @ W1[128,128]  (fp32 WMMA 16x16x4)
// 8 waves/block, wave w owns N-tile w. W1 staged once; each 16x128 x-tile is
// moved global->LDS by the Tensor Data Mover (TDM) with the row pad applied
// by the descriptor (pad 4 DWORDs every 128 DWORDs -> LDS row stride 132).
// 100000 == 6250*16 -> no partial M tiles.
// ---------------------------------------------------------------------------
__global__ __launch_bounds__(256) void gnn_gemm1_kernel(const float* __restrict__ x,
                                                        const float* __restrict__ W1,
                                                        float* __restrict__ h1) {
    // Row strides padded so WMMA operand reads from the two lane-halves
    // (rows k and k+2) land on disjoint LDS banks.
    __shared__ float lds_w[HID_CH][HID_CH + 8];   // stride 136 floats (16B mult)
    __shared__ float lds_x[16][IN_CH + 4];        // stride 132 floats (16B mult)

    const int tid  = threadIdx.x;
    const int wave = tid >> 5;
    const int lane = tid & 31;
    const int half = lane >> 4;   // 0: lanes 0-15, 1: lanes 16-31
    const int l15  = lane & 15;
    const int n0   = wave * 16;

    // Stage W1 into LDS (float4 both sides; LDS row stride is 16B-aligned).
    for (int i = tid * 4; i < IN_CH * HID_CH; i += 256 * 4) {
        const float4 w = *(const float4*)(W1 + i);
        *(float4*)&lds_w[i >> 7][i & 127] = w;
    }

    const int m_tiles = N_NODES / 16;   // 6250
    for (int mt = blockIdx.x; mt < m_tiles; mt += gridDim.x) {
        __syncthreads();   // all waves done reading previous x-tile
        if (wave == 0) {
            // --- TDM descriptor (ISA 8.3/8.4): 2D fp32 tile 128x16 -> LDS ---
            const unsigned           lds_base = (unsigned)(size_t)(&lds_x[0][0]);
            const unsigned long long ga = (unsigned long long)(size_t)(x + (size_t)mt * 16 * IN_CH);
            v4u g0;
            g0.x = 1u;                                   // count=1, user mode
            g0.y = lds_base;                             // lds_addr (bytes)
            g0.z = (unsigned)(ga & 0xFFFFFFFFu);         // global_addr[31:0]
            g0.w = (unsigned)((ga >> 32) & 0x01FFFFFFu)  // global_addr[56:32]
                 | (2u << 30);                           // type=2 ("image")
            v8i g1;
            g1[0] = (int)((2u << 16)     // data_size = 4 bytes
                        | (1u << 20)     // pad_enable
                        | (6u << 22)     // pad_interval: every 128 DWORDs
                        | (3u << 25));   // pad_amount: 4 DWORDs
            g1[1] = (int)(128u << 16);   // tensor_dim0[15:0] = 128
            g1[2] = (int)(16u << 16);    // tensor_dim1[15:0] = 16
            g1[3] = (int)(128u << 16);   // tile_dim0 = 128
            g1[4] = 16;                  // tile_dim1 = 16 (tile_dim2 = 0)
            g1[5] = 128;                 // tensor_dim0_stride[31:0] = 128
            g1[6] = 0;
            g1[7] = 0;
            const v4i g2 = {0, 0, 0, 0};           // dims 2/3 unused (2D tile)
            const v4i g3 = {0, 0, 0, 0};
            const v8i gx = {0, 0, 0, 0, 0, 0, 0, 0}; // extension group: unused
            __builtin_amdgcn_tensor_load_to_lds(g0, g1, g2, g3, gx, 0);
            __builtin_amdgcn_s_wait_tensorcnt(0);
        }
        __syncthreads();   // x-tile published to all waves

        v8f acc = {};
        for (int k0 = 0; k0 < IN_CH; k0 += 4) {
            const int ka = k0 + 2 * half;
            v2f a, b;
            // A 16x4: lane holds row M=l15, K = ka, ka+1 (ISA 7.12.2)
            a.x = lds_x[l15][ka];
            a.y = lds_x[l15][ka + 1];
            // B 4x16: VGPR v holds row K = 2*half + v, cols striped on lanes
            b.x = lds_w[ka][n0 + l15];
            b.y = lds_w[ka + 1][n0 + l15];
            acc = __builtin_amdgcn_wmma_f32_16x16x4_f32(false, a, false, b,
                                                        (short)0, acc, false, false);
        }
        // C/D: VGPR r -> (M = r + 8*half, N = l15)
        #pragma unroll
        for (int r = 0; r < 8; ++r)
            h1[(size_t)(mt * 16 + r + 8 * half) * HID_CH + n0 + l15] = acc[r];
    }
}

// ---------------------------------------------------------------------------
// GEMM2: h3[100000,64] = relu(h2[100000,128]) @ W2[128,64]
// 8 waves/block: waves 0-3 -> M-tile A (N-tiles 0-3), waves 4-7 -> M-tile B.
// relu fused into the (float4-vectorized) LDS staging.
// ---------------------------------------------------------------------------
__global__ __launch_bounds__(256) void gnn_gemm2_kernel(const float* __restrict__ h2,
                                                        const float* __restrict__ W2,
                                                        float* __restrict__ h3) {
    __shared__ float lds_w[HID_CH][OUT_CH + 8];   // stride 72 floats (16B mult)
    __shared__ float lds_x[32][IN_CH + 4];        // stride 132 floats (16B mult)

    const int tid  = threadIdx.x;
    const int wave = tid >> 5;
    const int lane = tid & 31;
    const int half = lane >> 4;
    const int l15  = lane & 15;
    const int sub  = wave >> 2;        // which of the 2 M-tiles
    const int n0   = (wave & 3) * 16;  // N-tile

    for (int i = tid * 4; i < HID_CH * OUT_CH; i += 256 * 4) {
        const float4 w = *(const float4*)(W2 + i);
        *(float4*)&lds_w[i >> 6][i & 63] = w;
    }

    const int mt_pairs = N_NODES / 32;  // 3125
    for (int mt = blockIdx.x; mt < mt_pairs; mt += gridDim.x) {
        __syncthreads();
        const int m0 = mt * 32;
        // 32x128 floats = 1024 float4; 4 per thread, relu fused.
        for (int i4 = tid; i4 < 32 * (IN_CH / 4); i4 += 256) {
            const int r = i4 >> 5, c = (i4 & 31) * 4;
            float4 v = *(const float4*)(h2 + (size_t)(m0 + r) * HID_CH + c);
            v.x = fmaxf(v.x, 0.f); v.y = fmaxf(v.y, 0.f);
            v.z = fmaxf(v.z, 0.f); v.w = fmaxf(v.w, 0.f);
            *(float4*)&lds_x[r][c] = v;
        }
        __syncthreads();

        const int mrow = sub * 16;
        v8f acc = {};
        for (int k0 = 0; k0 < HID_CH; k0 += 4) {
            const int ka = k0 + 2 * half;
            v2f a, b;
            a.x = lds_x[mrow + l15][ka];
            a.y = lds_x[mrow + l15][ka + 1];
            b.x = lds_w[ka][n0 + l15];
            b.y = lds_w[ka + 1][n0 + l15];
            acc = __builtin_amdgcn_wmma_f32_16x16x4_f32(false, a, false, b,
                                                        (short)0, acc, false, false);
        }
        #pragma unroll
        for (int r = 0; r < 8; ++r)
            h3[(size_t)(m0 + mrow + r + 8 * half) * OUT_CH + n0 + l15] = acc[r];
    }
}

// ---------------------------------------------------------------------------
// SpMM D=128: one wave per edge, 4 channels per lane. Destinations are
// L2-resident (51 MB < 192 MB L2) so the fp32 atomics stay on-chip.
// ---------------------------------------------------------------------------
__global__ __launch_bounds__(256) void gnn_spmm128_kernel(const float* __restrict__ h,
                                                          const float* __restrict__ vals,
                                                          const int* __restrict__ src,
                                                          const int* __restrict__ dst,
                                                          float* __restrict__ out) {
    const int e = (blockIdx.x * 256 + (int)threadIdx.x) >> 5;
    if (e >= N_EDGES) return;
    const int lane = threadIdx.x & 31;
    const int s = src[e];
    const int d = dst[e];
    const float v = vals[e];
    const float4 m = *(const float4*)(h + (size_t)s * 128 + lane * 4);
    float* o = out + (size_t)d * 128 + lane * 4;
    atomic_add_f32(o + 0, v * m.x);
    atomic_add_f32(o + 1, v * m.y);
    atomic_add_f32(o + 2, v * m.z);
    atomic_add_f32(o + 3, v * m.w);
}

// SpMM D=64: one wave per edge, 2 channels per lane.
__global__ __launch_bounds__(256) void gnn_spmm64_kernel(const float* __restrict__ h,
                                                         const float* __restrict__ vals,
                                                         const int* __restrict__ src,
                                                         const int* __restrict__ dst,
                                                         float* __restrict__ out) {
    const int e = (blockIdx.x * 256 + (int)threadIdx.x) >> 5;
    if (e >= N_EDGES) return;
    const int lane = threadIdx.x & 31;
    const int s = src[e];
    const int d = dst[e];
    const float v = vals[e];
    const float2 m = *(const float2*)(h + (size_t)s * 64 + lane * 2);
    float* o = out + (size_t)d * 64 + lane * 2;
    atomic_add_f32(o + 0, v * m.x);
    atomic_add_f32(o + 1, v * m.y);
}

// ---------------------------------------------------------------------------
extern "C" void kernel_launch(void* const* d_in, const int* in_sizes, int n_in,
                              void* d_out, int out_size, void* d_ws, size_t ws_size,
                              hipStream_t stream) {
    (void)in_sizes; (void)n_in; (void)out_size; (void)ws_size;

    const float* x     = (const float*)d_in[0];
    const float* W1    = (const float*)d_in[1];
    const float* W2    = (const float*)d_in[2];
    const float* evals = (const float*)d_in[3];
    const int*   eidx  = (const int*)d_in[4];
    const int*   esrc  = eidx;             // edge_index[0]
    const int*   edst  = eidx + N_EDGES;   // edge_index[1]
    float*       out   = (float*)d_out;

    float* h1 = (float*)d_ws;                                  // [100000,128] 51.2 MB
    float* h2 = h1 + (size_t)N_NODES * HID_CH;                 // [100000,128] 51.2 MB
    float* h3 = h2 + (size_t)N_NODES * HID_CH;                 // [100000, 64] 25.6 MB

    const int spmm_blocks = N_EDGES / 8;   // 8 waves (edges) per 256-thread block

    gnn_gemm1_kernel<<<2048, 256, 0, stream>>>(x, W1, h1);

    gnn_zero_kernel<<<2048, 256, 0, stream>>>((float4*)h2, (N_NODES * HID_CH) / 4);
    gnn_spmm128_kernel<<<spmm_blocks, 256, 0, stream>>>(h1, evals, esrc, edst, h2);

    gnn_gemm2_kernel<<<2048, 256, 0, stream>>>(h2, W2, h3);

    gnn_zero_kernel<<<2048, 256, 0, stream>>>((float4*)out, (N_NODES * OUT_CH) / 4);
    gnn_spmm64_kernel<<<spmm_blocks, 256, 0, stream>>>(h3, evals, esrc, edst, out);
}